// GAT_86749749444964
// MI455X (gfx1250) — compile-verified
//
#include <hip/hip_runtime.h>
#include <math.h>

// ---------------- problem constants ----------------
#define NN   100000      // nodes
#define EE   1600000     // edges (before self loops)
#define ETOT 1700000     // EE + NN (with self loops)
#define KIN  128         // input dim
#define C1   256         // H*HID
#define HEADS 4
#define NEG_SLOPE 0.2f

typedef __attribute__((ext_vector_type(2))) float v2f;
typedef __attribute__((ext_vector_type(8))) float v8f;

// ---------------- helpers ----------------
__device__ __forceinline__ void edge_sd(const int* __restrict__ ei, int e, int& s, int& d) {
  if (e < EE) { s = ei[e]; d = ei[EE + e]; }
  else        { s = e - EE; d = e - EE; }           // self loops appended
}

__device__ __forceinline__ float lrelu(float x) {
  return x >= 0.f ? x : NEG_SLOPE * x;
}

// float atomic max via signed/unsigned int monotonicity (works with -inf init)
__device__ __forceinline__ void atomicMaxF(float* addr, float val) {
  if (val >= 0.f) atomicMax((int*)addr, __float_as_int(val));
  else            atomicMin((unsigned int*)addr, __float_as_uint(val));
}

// ---------------- generic fill ----------------
__global__ void gat_fill(float* __restrict__ p, float v, int n) {
  int i = blockIdx.x * blockDim.x + threadIdx.x;
  if (i < n) p[i] = v;
}

// ---------------- GEMM1: h1[N,256] = x[N,128] @ W1[128,256] (fp32 WMMA) ----------------
// one wave -> 16 rows x 64 cols (4 accumulator tiles), K=128 in steps of 4
__global__ void gat_gemm1_wmma(const float* __restrict__ x, const float* __restrict__ W,
                               float* __restrict__ h1) {
  const int wid  = blockIdx.x * (blockDim.x >> 5) + (threadIdx.x >> 5);
  const int lane = threadIdx.x & 31;
  const int mtile = wid >> 2;          // 0..6249
  const int ngrp  = wid & 3;           // 0..3  (64 cols each)
  const int row   = mtile * 16 + (lane & 15);
  const int n0    = ngrp * 64;
  const int col   = n0 + (lane & 15);
  const int khalf = (lane >> 4) << 1;  // lanes 0-15 -> K+{0,1}; lanes 16-31 -> K+{2,3}

  v8f acc0 = {}, acc1 = {}, acc2 = {}, acc3 = {};
  const float* xrow = x + (size_t)row * KIN + khalf;

  for (int k0 = 0; k0 < KIN; k0 += 4) {
    v2f a = *(const v2f*)(xrow + k0);                  // contiguous K pair
    const float* wr0 = W + (size_t)(k0 + khalf) * C1;  // B row K (per half-wave)
    const float* wr1 = wr0 + C1;                       // B row K+1
    v2f b0, b1, b2, b3;
    b0.x = wr0[col +  0]; b0.y = wr1[col +  0];
    b1.x = wr0[col + 16]; b1.y = wr1[col + 16];
    b2.x = wr0[col + 32]; b2.y = wr1[col + 32];
    b3.x = wr0[col + 48]; b3.y = wr1[col + 48];
    acc0 = __builtin_amdgcn_wmma_f32_16x16x4_f32(false, a, false, b0, (short)0, acc0, false, false);
    acc1 = __builtin_amdgcn_wmma_f32_16x16x4_f32(false, a, false, b1, (short)0, acc1, false, false);
    acc2 = __builtin_amdgcn_wmma_f32_16x16x4_f32(false, a, false, b2, (short)0, acc2, false, false);
    acc3 = __builtin_amdgcn_wmma_f32_16x16x4_f32(false, a, false, b3, (short)0, acc3, false, false);
  }

  // C/D layout: VGPR r -> M = r (lanes 0-15) / 8+r (lanes 16-31), N = lane&15
  const int mrow = mtile * 16 + (lane >> 4) * 8;
#pragma unroll
  for (int r = 0; r < 8; ++r) {
    size_t base = (size_t)(mrow + r) * C1;
    h1[base + col +  0] = acc0[r];
    h1[base + col + 16] = acc1[r];
    h1[base + col + 32] = acc2[r];
    h1[base + col + 48] = acc3[r];
  }
}

// ---------------- per-node attention scores, layer 1 ----------------
// wave per node; lanes 0-7 head0, 8-15 head1, ... (8 channels per lane)
__global__ void gat_scores1(const float* __restrict__ h1, const float* __restrict__ a_s,
                            const float* __restrict__ a_d, float* __restrict__ ssrc,
                            float* __restrict__ sdst) {
  int node = blockIdx.x * (blockDim.x >> 5) + (threadIdx.x >> 5);
  int lane = threadIdx.x & 31;
  int head = lane >> 3;
  int coff = head * 64 + (lane & 7) * 8;
  const float* hp = h1 + (size_t)node * C1 + coff;
  float ps = 0.f, pd = 0.f;
#pragma unroll
  for (int t = 0; t < 8; ++t) {
    float v = hp[t];
    ps += v * a_s[coff + t];
    pd += v * a_d[coff + t];
  }
#pragma unroll
  for (int off = 1; off < 8; off <<= 1) {
    ps += __shfl_xor(ps, off, 32);
    pd += __shfl_xor(pd, off, 32);
  }
  if ((lane & 7) == 0) {
    ssrc[node * HEADS + head] = ps;
    sdst[node * HEADS + head] = pd;
  }
}

// ---------------- edge phase layer 1 ----------------
__global__ void gat_edge_max1(const int* __restrict__ ei, const float* __restrict__ ssrc,
                              const float* __restrict__ sdst, float* __restrict__ m) {
  int idx = blockIdx.x * blockDim.x + threadIdx.x;
  if (idx >= ETOT * HEADS) return;
  int e = idx >> 2, h = idx & 3;
  int s, d; edge_sd(ei, e, s, d);
  float sc = lrelu(ssrc[s * HEADS + h] + sdst[d * HEADS + h]);
  atomicMaxF(&m[d * HEADS + h], sc);
}

__global__ void gat_fix_m(float* __restrict__ m, int n) {
  int i = blockIdx.x * blockDim.x + threadIdx.x;
  if (i >= n) return;
  float v = m[i];
  if (!isfinite(v)) m[i] = 0.f;
}

__global__ void gat_edge_expsum1(const int* __restrict__ ei, const float* __restrict__ ssrc,
                                 const float* __restrict__ sdst, const float* __restrict__ m,
                                 float* __restrict__ ex, float* __restrict__ den) {
  int idx = blockIdx.x * blockDim.x + threadIdx.x;
  if (idx >= ETOT * HEADS) return;
  int e = idx >> 2, h = idx & 3;
  int s, d; edge_sd(ei, e, s, d);
  float sc = lrelu(ssrc[s * HEADS + h] + sdst[d * HEADS + h]);
  float v = expf(sc - m[d * HEADS + h]);
  ex[(size_t)e * HEADS + h] = v;
  atomicAdd(&den[d * HEADS + h], v);
}

// wave per edge: alpha-weighted scatter of 256 channels (8 coalesced chunks)
__global__ void gat_aggregate1(const int* __restrict__ ei, const float* __restrict__ ex,
                               const float* __restrict__ den, const float* __restrict__ h1,
                               float* __restrict__ out1) {
  int e = blockIdx.x * (blockDim.x >> 5) + (threadIdx.x >> 5);
  int lane = threadIdx.x & 31;
  if (e >= ETOT) return;
  int s, d; edge_sd(ei, e, s, d);
  float alpha4 = 0.f;
  if (lane < HEADS)
    alpha4 = ex[(size_t)e * HEADS + lane] / den[(size_t)d * HEADS + lane];
  const float* hs = h1 + (size_t)s * C1;
  float* od = out1 + (size_t)d * C1;
#pragma unroll
  for (int it = 0; it < 8; ++it) {
    int c = it * 32 + lane;
    float alpha = __shfl(alpha4, c >> 6, 32);
    atomicAdd(od + c, alpha * hs[c]);
  }
}

__global__ void gat_elu_bias(float* __restrict__ out1, const float* __restrict__ b) {
  int i = blockIdx.x * blockDim.x + threadIdx.x;   // exactly N*256 threads
  float v = out1[i] + b[i & (C1 - 1)];
  out1[i] = v > 0.f ? v : (expf(v) - 1.f);
}

// ---------------- layer 2: g[N,2] = h[N,256] @ W2[256,2] + scores ----------------
__global__ void gat_gemm2_scores(const float* __restrict__ h, const float* __restrict__ W2,
                                 const float* __restrict__ a2s, const float* __restrict__ a2d,
                                 float* __restrict__ g2, float* __restrict__ ssrc2,
                                 float* __restrict__ sdst2) {
  int node = blockIdx.x * (blockDim.x >> 5) + (threadIdx.x >> 5);
  int lane = threadIdx.x & 31;
  const float* hp = h + (size_t)node * C1 + lane * 8;
  float g0 = 0.f, g1 = 0.f;
#pragma unroll
  for (int t = 0; t < 8; ++t) {
    float v = hp[t];
    int c = lane * 8 + t;
    g0 += v * W2[c * 2 + 0];
    g1 += v * W2[c * 2 + 1];
  }
#pragma unroll
  for (int off = 1; off < 32; off <<= 1) {
    g0 += __shfl_xor(g0, off, 32);
    g1 += __shfl_xor(g1, off, 32);
  }
  if (lane == 0) {
    g2[node * 2 + 0] = g0;
    g2[node * 2 + 1] = g1;
    ssrc2[node] = g0 * a2s[0] + g1 * a2s[1];
    sdst2[node] = g0 * a2d[0] + g1 * a2d[1];
  }
}

__global__ void gat_edge_max2(const int* __restrict__ ei, const float* __restrict__ ssrc,
                              const float* __restrict__ sdst, float* __restrict__ m) {
  int e = blockIdx.x * blockDim.x + threadIdx.x;
  if (e >= ETOT) return;
  int s, d; edge_sd(ei, e, s, d);
  atomicMaxF(&m[d], lrelu(ssrc[s] + sdst[d]));
}

__global__ void gat_edge_expsum2(const int* __restrict__ ei, const float* __restrict__ ssrc,
                                 const float* __restrict__ sdst, const float* __restrict__ m,
                                 float* __restrict__ ex, float* __restrict__ den) {
  int e = blockIdx.x * blockDim.x + threadIdx.x;
  if (e >= ETOT) return;
  int s, d; edge_sd(ei, e, s, d);
  float v = expf(lrelu(ssrc[s] + sdst[d]) - m[d]);
  ex[e] = v;
  atomicAdd(&den[d], v);
}

__global__ void gat_aggregate2(const int* __restrict__ ei, const float* __restrict__ ex,
                               const float* __restrict__ den, const float* __restrict__ g2,
                               float* __restrict__ out) {
  int e = blockIdx.x * blockDim.x + threadIdx.x;
  if (e >= ETOT) return;
  int s, d; edge_sd(ei, e, s, d);
  float alpha = ex[e] / den[d];
  atomicAdd(&out[d * 2 + 0], alpha * g2[s * 2 + 0]);
  atomicAdd(&out[d * 2 + 1], alpha * g2[s * 2 + 1]);
}

__global__ void gat_bias2(float* __restrict__ out, const float* __restrict__ b2) {
  int i = blockIdx.x * blockDim.x + threadIdx.x;
  if (i >= 2 * NN) return;
  out[i] += b2[i & 1];
}

// ---------------- launch ----------------
extern "C" void kernel_launch(void* const* d_in, const int* in_sizes, int n_in,
                              void* d_out, int out_size, void* d_ws, size_t ws_size,
                              hipStream_t stream) {
  (void)in_sizes; (void)n_in; (void)out_size; (void)ws_size;
  const float* x   = (const float*)d_in[0];
  const int*   ei  = (const int*)  d_in[1];
  const float* W1  = (const float*)d_in[2];
  const float* a1s = (const float*)d_in[3];
  const float* a1d = (const float*)d_in[4];
  const float* b1  = (const float*)d_in[5];
  const float* W2  = (const float*)d_in[6];
  const float* a2s = (const float*)d_in[7];
  const float* a2d = (const float*)d_in[8];
  const float* b2  = (const float*)d_in[9];
  float* out = (float*)d_out;

  // workspace carve-up (floats)
  float* ws = (float*)d_ws;
  float* h1    = ws;  ws += (size_t)NN * C1;        // 25.6M
  float* out1  = ws;  ws += (size_t)NN * C1;        // 25.6M
  float* ssrc1 = ws;  ws += (size_t)NN * HEADS;
  float* sdst1 = ws;  ws += (size_t)NN * HEADS;
  float* m1    = ws;  ws += (size_t)NN * HEADS;
  float* den1  = ws;  ws += (size_t)NN * HEADS;
  float* ex1   = ws;  ws += (size_t)ETOT * HEADS;   // 6.8M
  float* g2    = ws;  ws += (size_t)NN * 2;
  float* ssrc2 = ws;  ws += (size_t)NN;
  float* sdst2 = ws;  ws += (size_t)NN;
  float* m2    = ws;  ws += (size_t)NN;
  float* den2  = ws;  ws += (size_t)NN;
  float* ex2   = ws;  ws += (size_t)ETOT;

  const int T = 256;
  auto blk = [](long long n, int t) { return (unsigned)((n + t - 1) / t); };

  // init accumulators / segment buffers
  gat_fill<<<blk((long long)NN * C1, T), T, 0, stream>>>(out1, 0.f, NN * C1);
  gat_fill<<<blk(NN * HEADS, T), T, 0, stream>>>(den1, 0.f, NN * HEADS);
  gat_fill<<<blk(NN * HEADS, T), T, 0, stream>>>(m1, -INFINITY, NN * HEADS);
  gat_fill<<<blk(NN, T), T, 0, stream>>>(den2, 0.f, NN);
  gat_fill<<<blk(NN, T), T, 0, stream>>>(m2, -INFINITY, NN);
  gat_fill<<<blk(2 * NN, T), T, 0, stream>>>(out, 0.f, 2 * NN);

  // layer 1
  gat_gemm1_wmma<<<(NN / 16) * 4 / 8, T, 0, stream>>>(x, W1, h1);          // 3125 blocks
  gat_scores1<<<NN / 8, T, 0, stream>>>(h1, a1s, a1d, ssrc1, sdst1);       // 12500 blocks
  gat_edge_max1<<<blk((long long)ETOT * HEADS, T), T, 0, stream>>>(ei, ssrc1, sdst1, m1);
  gat_fix_m<<<blk(NN * HEADS, T), T, 0, stream>>>(m1, NN * HEADS);
  gat_edge_expsum1<<<blk((long long)ETOT * HEADS, T), T, 0, stream>>>(ei, ssrc1, sdst1, m1, ex1, den1);
  gat_aggregate1<<<ETOT / 8, T, 0, stream>>>(ei, ex1, den1, h1, out1);     // 212500 blocks
  gat_elu_bias<<<blk((long long)NN * C1, T), T, 0, stream>>>(out1, b1);

  // layer 2
  gat_gemm2_scores<<<NN / 8, T, 0, stream>>>(out1, W2, a2s, a2d, g2, ssrc2, sdst2);
  gat_edge_max2<<<blk(ETOT, T), T, 0, stream>>>(ei, ssrc2, sdst2, m2);
  gat_fix_m<<<blk(NN, T), T, 0, stream>>>(m2, NN);
  gat_edge_expsum2<<<blk(ETOT, T), T, 0, stream>>>(ei, ssrc2, sdst2, m2, ex2, den2);
  gat_aggregate2<<<blk(ETOT, T), T, 0, stream>>>(ei, ex2, den2, g2, out);
  gat_bias2<<<blk(2 * NN, T), T, 0, stream>>>(out, b2);
}